// HyperMod_77644418777859
// MI455X (gfx1250) — compile-verified
//
#include <hip/hip_runtime.h>
#include <stddef.h>
#include <stdint.h>

#define HDIM 256
#define WSTRIDE 260   // 256 + 4 pad dwords (TDM pad_interval=256dw, pad_amount=4dw)
#define ASTRIDE 36    // 32 + 4 pad dwords  (TDM pad_interval=32dw,  pad_amount=4dw)

typedef float v2f __attribute__((ext_vector_type(2)));
typedef float v8f __attribute__((ext_vector_type(8)));
typedef unsigned int u32x4 __attribute__((ext_vector_type(4)));
typedef int i32x4 __attribute__((ext_vector_type(4)));
typedef int i32x8 __attribute__((ext_vector_type(8)));

// -------------------------------------------------------------------------
// TDM 2D tile load, fp32 elements, with LDS row padding.
//   tile = tile_w x tile_h elements starting at gptr; row stride in memory =
//   row_stride_elems; rows >= tensor_h (relative to gptr) read as zero.
//   pad codes per ISA: interval 0=2dw..7=256dw ; amount 0=1dw..127=128dw
// -------------------------------------------------------------------------
__device__ __forceinline__ void tdm_load_tile_f32(
    uint32_t lds_byte_addr, const float* gptr,
    int tile_w, int tile_h, int tensor_h, int row_stride_elems,
    int pad_interval_code, int pad_amount_code)
{
  uint64_t ga = (uint64_t)(uintptr_t)gptr;
  u32x4 g0;
  g0[0] = 1u;                                            // count=1, user D#
  g0[1] = lds_byte_addr;                                 // lds_addr
  g0[2] = (uint32_t)ga;                                  // global_addr[31:0]
  g0[3] = (uint32_t)((ga >> 32) & 0x01FFFFFFu) | (2u << 30); // addr[56:32] | type=2
  i32x8 g1;
  g1[0] = (2 << 16) | (1 << 20) |                        // data_size=4B, pad_enable
          (pad_interval_code << 22) | (pad_amount_code << 25);
  g1[1] = (tile_w & 0xFFFF) << 16;                       // tensor_dim0[15:0]
  g1[2] = ((tile_w >> 16) & 0xFFFF) | ((tensor_h & 0xFFFF) << 16); // dim0 hi | dim1 lo
  g1[3] = ((tensor_h >> 16) & 0xFFFF) | ((tile_w & 0xFFFF) << 16); // dim1 hi | tile_dim0
  g1[4] = (tile_h & 0xFFFF);                             // tile_dim1 (tile_dim2=0)
  g1[5] = row_stride_elems;                              // tensor_dim0_stride[31:0]
  g1[6] = 0;
  g1[7] = 0;
  i32x4 z4 = {0, 0, 0, 0};
  i32x8 z8 = {0, 0, 0, 0, 0, 0, 0, 0};
  __builtin_amdgcn_tensor_load_to_lds(g0, g1, z4, z4, z8, 0);
}

// -------------------------------------------------------------------------
// out[m][n] = relu( sum_k A[m][k]*W[n][k] + bias[n] ) * rowscale[m]
// Grid: x = ceil(M/128), y = 4 (64 cols/block); block = 256 threads (8 waves).
// W strip staged once via TDM; A tiles (128x32) double-buffered via TDM.
// Each wave computes a 16(M) x 64(N) strip with V_WMMA_F32_16X16X4_F32.
// -------------------------------------------------------------------------
__global__ __launch_bounds__(256) void gemm_relu_scale_wmma(
    const float* __restrict__ A, const float* __restrict__ W,
    const float* __restrict__ bias, const float* __restrict__ rowscale,
    float* __restrict__ out, int M)
{
  __shared__ float shW[64 * WSTRIDE];        // 66,560 B
  __shared__ float shA[2][128 * ASTRIDE];    // 2 x 18,432 B

  const int wave = threadIdx.x >> 5;
  const int lane = threadIdx.x & 31;
  const int half = lane >> 4;                // 0: lanes 0-15, 1: lanes 16-31
  const int l15  = lane & 15;
  const int m_base = blockIdx.x * 128;
  const int n_base = blockIdx.y * 64;

  if (wave == 0) {
    // W strip: 64 rows x 256 k, LDS row stride 260 (pad 4dw / 256dw)
    tdm_load_tile_f32((uint32_t)(uintptr_t)&shW[0],
                      W + (size_t)n_base * HDIM,
                      HDIM, 64, 64, HDIM, /*interval 256dw*/7, /*amount 4dw*/3);
    // A chunk 0: 128 rows x 32 k, LDS row stride 36 (pad 4dw / 32dw)
    tdm_load_tile_f32((uint32_t)(uintptr_t)&shA[0][0],
                      A + (size_t)m_base * HDIM,
                      32, 128, M - m_base, HDIM, /*interval 32dw*/4, /*amount 4dw*/3);
  }

  v8f acc[4];
#pragma unroll
  for (int t = 0; t < 4; ++t)
#pragma unroll
    for (int j = 0; j < 8; ++j) acc[t][j] = 0.0f;

  const int NCHUNK = HDIM / 32;              // 8
  for (int c = 0; c < NCHUNK; ++c) {
    if (wave == 0) {
      __builtin_amdgcn_s_wait_tensorcnt(0);  // chunk c (and W) resident in LDS
      if (c + 1 < NCHUNK)                    // prefetch next chunk while computing
        tdm_load_tile_f32((uint32_t)(uintptr_t)&shA[(c + 1) & 1][0],
                          A + (size_t)m_base * HDIM + (c + 1) * 32,
                          32, 128, M - m_base, HDIM, 4, 3);
    }
    __syncthreads();

    const float* ap = &shA[c & 1][(wave * 16 + l15) * ASTRIDE + half * 2];
    const float* b0 = &shW[(l15 +  0) * WSTRIDE + c * 32 + half * 2];
    const float* b1 = &shW[(l15 + 16) * WSTRIDE + c * 32 + half * 2];
    const float* b2 = &shW[(l15 + 32) * WSTRIDE + c * 32 + half * 2];
    const float* b3 = &shW[(l15 + 48) * WSTRIDE + c * 32 + half * 2];

#pragma unroll
    for (int kk = 0; kk < 32; kk += 4) {
      v2f a  = *(const v2f*)(ap + kk);
      v2f f0 = *(const v2f*)(b0 + kk);
      v2f f1 = *(const v2f*)(b1 + kk);
      v2f f2 = *(const v2f*)(b2 + kk);
      v2f f3 = *(const v2f*)(b3 + kk);
      acc[0] = __builtin_amdgcn_wmma_f32_16x16x4_f32(false, a, false, f0, (short)0, acc[0], false, false);
      acc[1] = __builtin_amdgcn_wmma_f32_16x16x4_f32(false, a, false, f1, (short)0, acc[1], false, false);
      acc[2] = __builtin_amdgcn_wmma_f32_16x16x4_f32(false, a, false, f2, (short)0, acc[2], false, false);
      acc[3] = __builtin_amdgcn_wmma_f32_16x16x4_f32(false, a, false, f3, (short)0, acc[3], false, false);
    }
    __syncthreads();
  }

  float bn[4];
#pragma unroll
  for (int t = 0; t < 4; ++t) bn[t] = bias[n_base + t * 16 + l15];

  // C/D layout: VGPR j holds rows m0+j (lanes 0-15) and m0+8+j (lanes 16-31)
#pragma unroll
  for (int j = 0; j < 8; ++j) {
    int row = m_base + wave * 16 + half * 8 + j;
    if (row < M) {
      float rs = rowscale[row];
#pragma unroll
      for (int t = 0; t < 4; ++t) {
        float val = acc[t][j] + bn[t];
        val = val > 0.0f ? val : 0.0f;
        out[(size_t)row * HDIM + n_base + t * 16 + l15] = val * rs;
      }
    }
  }
}

// -------------------------------------------------------------------------
// Elementwise helpers (float4 vectorized; 64 float4 per 256-wide row)
// -------------------------------------------------------------------------
__global__ __launch_bounds__(256) void copy_k(const float4* __restrict__ src,
                                              float4* __restrict__ dst, int n4)
{
  int i = blockIdx.x * 256 + threadIdx.x;
  if (i < n4) dst[i] = src[i];
}

__global__ __launch_bounds__(256) void rowmul_k(const float4* __restrict__ src,
                                                const float* __restrict__ w,
                                                float4* __restrict__ dst, int n4)
{
  int i = blockIdx.x * 256 + threadIdx.x;
  if (i < n4) {
    float s = w[i >> 6];
    float4 x = src[i];
    x.x *= s; x.y *= s; x.z *= s; x.w *= s;
    dst[i] = x;
  }
}

__global__ __launch_bounds__(256) void rowdiv_k(float4* __restrict__ buf,
                                                const float* __restrict__ s, int n4)
{
  int i = blockIdx.x * 256 + threadIdx.x;
  if (i < n4) {
    float inv = 1.0f / s[i >> 6];
    float4 x = buf[i];
    x.x *= inv; x.y *= inv; x.z *= inv; x.w *= inv;
    buf[i] = x;
  }
}

// -------------------------------------------------------------------------
// Scatter: dst[sidx[p]][:] += src[gidx[p]][:] * pw[p]
// 64 lanes per pair, float4 gather + 4 hardware f32 atomics per lane.
// -------------------------------------------------------------------------
__global__ __launch_bounds__(256) void scatter_add_k(
    const float* __restrict__ src, const int* __restrict__ gidx,
    const int* __restrict__ sidx, const float* __restrict__ pw,
    float* __restrict__ dst, int P)
{
  int p = blockIdx.x * 4 + (threadIdx.x >> 6);
  if (p >= P) return;
  int t = threadIdx.x & 63;
  int g = gidx[p];
  int s = sidx[p];
  float w = pw[p];
  float4 x = ((const float4*)(src + (size_t)g * HDIM))[t];
  float* d = dst + (size_t)s * HDIM + t * 4;
  __hip_atomic_fetch_add(d + 0, x.x * w, __ATOMIC_RELAXED, __HIP_MEMORY_SCOPE_AGENT);
  __hip_atomic_fetch_add(d + 1, x.y * w, __ATOMIC_RELAXED, __HIP_MEMORY_SCOPE_AGENT);
  __hip_atomic_fetch_add(d + 2, x.z * w, __ATOMIC_RELAXED, __HIP_MEMORY_SCOPE_AGENT);
  __hip_atomic_fetch_add(d + 3, x.w * w, __ATOMIC_RELAXED, __HIP_MEMORY_SCOPE_AGENT);
}

// -------------------------------------------------------------------------
extern "C" void kernel_launch(void* const* d_in, const int* in_sizes, int n_in,
                              void* d_out, int out_size, void* d_ws, size_t ws_size,
                              hipStream_t stream)
{
  const float* v            = (const float*)d_in[0];
  const float* e            = (const float*)d_in[1];
  const float* W_v2e_w      = (const float*)d_in[2];
  const float* W_v2e_b      = (const float*)d_in[3];
  const float* W_e2v_w      = (const float*)d_in[4];
  const float* W_e2v_b      = (const float*)d_in[5];
  const float* n_weight     = (const float*)d_in[6];
  const float* e_weight     = (const float*)d_in[7];
  const float* n_reg_weight = (const float*)d_in[8];
  const float* e_reg_weight = (const float*)d_in[9];
  const float* e_reg_sum    = (const float*)d_in[10];
  const float* n_reg_sum    = (const float*)d_in[11];
  const int*   pairs_v      = (const int*)d_in[12];
  const int*   eidx         = (const int*)d_in[13];
  const int*   pairs_e      = (const int*)d_in[14];
  const int*   vidx         = (const int*)d_in[15];

  const int NV = in_sizes[0] / HDIM;
  const int NE = in_sizes[1] / HDIM;
  const int P  = in_sizes[12];

  float* v_out = (float*)d_out;                     // [NV,256]
  float* e_out = (float*)d_out + (size_t)NV * HDIM; // [NE,256]
  float* ve_s  = (float*)d_ws;                      // [NV,256] scratch
  float* ev_s  = ve_s + (size_t)NV * HDIM;          // [NE,256] scratch

  const int nv4 = NV * HDIM / 4;
  const int ne4 = NE * HDIM / 4;
  dim3 blk(256);

  // 1. ve_s = relu(v @ W_v2e^T + b) * n_weight
  gemm_relu_scale_wmma<<<dim3((NV + 127) / 128, 4), blk, 0, stream>>>(
      v, W_v2e_w, W_v2e_b, n_weight, ve_s, NV);
  // 2. e_out = e   (accumulator lives in d_out)
  copy_k<<<(ne4 + 255) / 256, blk, 0, stream>>>((const float4*)e, (float4*)e_out, ne4);
  // 3. e_out[eidx[p]] += ve_s[pairs_v[p]] * n_reg_weight[p]
  scatter_add_k<<<(P + 3) / 4, blk, 0, stream>>>(ve_s, pairs_v, eidx, n_reg_weight, e_out, P);
  // 4. e_out /= e_reg_sum
  rowdiv_k<<<(ne4 + 255) / 256, blk, 0, stream>>>((float4*)e_out, e_reg_sum, ne4);
  // 5. ev_s = relu(e_out @ W_e2v^T + b) * e_weight
  gemm_relu_scale_wmma<<<dim3((NE + 127) / 128, 4), blk, 0, stream>>>(
      e_out, W_e2v_w, W_e2v_b, e_weight, ev_s, NE);
  // 6. v_out = v * n_weight
  rowmul_k<<<(nv4 + 255) / 256, blk, 0, stream>>>((const float4*)v, n_weight, (float4*)v_out, nv4);
  // 7. v_out[vidx[p]] += ev_s[pairs_e[p]] * e_reg_weight[p]
  scatter_add_k<<<(P + 3) / 4, blk, 0, stream>>>(ev_s, pairs_e, vidx, e_reg_weight, v_out, P);
  // 8. v_out /= n_reg_sum
  rowdiv_k<<<(nv4 + 255) / 256, blk, 0, stream>>>((float4*)v_out, n_reg_sum, nv4);
}